// Circuit_21732534518253
// MI455X (gfx1250) — compile-verified
//
#include <hip/hip_runtime.h>
#include <hip/hip_bf16.h>
#include <math.h>

// ---------------------------------------------------------------------------
// Photonic circuit RK4 integrator for MI455X (gfx1250).
//   f(A) = i(omega + nl2*|A|^2) .* A + A @ T2^T
// Complex GEMM done as real (16x64)@(64x64) via V_WMMA_F32_16X16X4_F32.
// RK4 state + nonlinearity kept in registers in WMMA D-fragment layout.
// ---------------------------------------------------------------------------

typedef __attribute__((ext_vector_type(2))) float v2f;
typedef __attribute__((ext_vector_type(8))) float v8f;

#define N_MODES 32
#define IN_MODES 24
#define LAMBDA_P 0.1f
#define EVAL_PTS 200

// ------------------------- complex helpers (float2) ------------------------
__device__ __forceinline__ float2 cmul(float2 a, float2 b) {
    return make_float2(a.x * b.x - a.y * b.y, a.x * b.y + a.y * b.x);
}
__device__ __forceinline__ float2 cconj(float2 a) { return make_float2(a.x, -a.y); }
__device__ __forceinline__ float2 cadd(float2 a, float2 b) { return make_float2(a.x + b.x, a.y + b.y); }
__device__ __forceinline__ float2 csub(float2 a, float2 b) { return make_float2(a.x - b.x, a.y - b.y); }

// ===========================================================================
// Kernel 1: build W(64x64) = [[Sr, Si], [-Si, Sr]], S = T2^T, from params.
// One block of 1024 threads; thread t -> (i = t>>5, j = t&31).
// ===========================================================================
__global__ __launch_bounds__(1024) void setup_kernel(const float* __restrict__ params,
                                                     const float* __restrict__ kappa,
                                                     float* __restrict__ Wg) {
    __shared__ float2 sH[32][33];
    __shared__ float2 sV[32][33];
    __shared__ float2 sU[32][33];
    __shared__ float2 sG[32][33];
    __shared__ float2 sM[32][33];
    __shared__ float2 sI[32][33];
    __shared__ int sPiv;

    const int tid = threadIdx.x;
    const int i = tid >> 5;
    const int j = tid & 31;

    // ---- init Hermitian H and V = I ----
    float2 h;
    if (i < j) {
        int off = 31 * i - (i * (i - 1)) / 2;
        int idx = off + (j - i - 1);
        h = make_float2(params[idx], params[496 + idx]);
    } else if (i > j) {
        int off = 31 * j - (j * (j - 1)) / 2;
        int idx = off + (i - j - 1);
        h = make_float2(params[idx], -params[496 + idx]);
    } else {
        if (i < 31) {
            h = make_float2(params[992 + i], 0.0f);
        } else {
            float ssum = 0.0f;
            for (int r = 0; r < 31; ++r) ssum += params[992 + r];
            h = make_float2(-ssum, 0.0f);
        }
    }
    sH[i][j] = h;
    sV[i][j] = make_float2((i == j) ? 1.0f : 0.0f, 0.0f);

    // ---- cyclic complex Jacobi eigensolver (20 sweeps) ----
    for (int sweep = 0; sweep < 20; ++sweep) {
        for (int p = 0; p < 31; ++p) {
            for (int q = p + 1; q < 32; ++q) {
                __syncthreads();  // previous rotation's writes visible
                float x = sH[p][q].x;
                float y = sH[p][q].y;
                float app = sH[p][p].x;
                float aqq = sH[q][q].x;
                float r2 = x * x + y * y;
                bool doit = (r2 > 1e-24f);
                float c = 1.0f, s = 0.0f, er = 1.0f, ei = 0.0f;
                if (doit) {
                    float r = sqrtf(r2);
                    er = x / r;
                    ei = y / r;
                    float tau = (aqq - app) / (2.0f * r);
                    float sgn = (tau >= 0.0f) ? 1.0f : -1.0f;
                    float tt = sgn / (fabsf(tau) + sqrtf(1.0f + tau * tau));
                    c = 1.0f / sqrtf(1.0f + tt * tt);
                    s = tt * c;
                }
                __syncthreads();  // all threads read before any write
                if (tid < 32 && doit) {
                    // column update: H <- H*G, V <- V*G
                    int k = tid;
                    float2 hp = sH[k][p], hq = sH[k][q];
                    float2 ebq = make_float2(er * hq.x + ei * hq.y, er * hq.y - ei * hq.x);
                    sH[k][p] = make_float2(c * hp.x - s * ebq.x, c * hp.y - s * ebq.y);
                    float2 efp = make_float2(er * hp.x - ei * hp.y, er * hp.y + ei * hp.x);
                    sH[k][q] = make_float2(s * efp.x + c * hq.x, s * efp.y + c * hq.y);

                    float2 vp = sV[k][p], vq = sV[k][q];
                    float2 vbq = make_float2(er * vq.x + ei * vq.y, er * vq.y - ei * vq.x);
                    sV[k][p] = make_float2(c * vp.x - s * vbq.x, c * vp.y - s * vbq.y);
                    float2 vfp = make_float2(er * vp.x - ei * vp.y, er * vp.y + ei * vp.x);
                    sV[k][q] = make_float2(s * vfp.x + c * vq.x, s * vfp.y + c * vq.y);
                }
                __syncthreads();
                if (tid < 32 && doit) {
                    // row update: H <- G^H * H
                    int k = tid;
                    float2 hp = sH[p][k], hq = sH[q][k];
                    float2 efq = make_float2(er * hq.x - ei * hq.y, er * hq.y + ei * hq.x);
                    sH[p][k] = make_float2(c * hp.x - s * efq.x, c * hp.y - s * efq.y);
                    float2 ebp = make_float2(er * hp.x + ei * hp.y, er * hp.y - ei * hp.x);
                    sH[q][k] = make_float2(s * ebp.x + c * hq.x, s * ebp.y + c * hq.y);
                }
            }
        }
    }
    __syncthreads();

    // ---- T1 = V * diag(e^{i w}) into sM ----
    float wj = sH[j][j].x;
    float2 ph = make_float2(cosf(wj), sinf(wj));
    sM[i][j] = cmul(sV[i][j], ph);
    __syncthreads();

    // ---- U = T1 @ V^H ----
    {
        float2 acc = make_float2(0.0f, 0.0f);
        for (int k = 0; k < 32; ++k) acc = cadd(acc, cmul(sM[i][k], cconj(sV[j][k])));
        sU[i][j] = acc;
    }
    __syncthreads();

    // ---- G = U^T @ U (plain transpose, no conjugate) ----
    float2 gij;
    {
        float2 acc = make_float2(0.0f, 0.0f);
        for (int k = 0; k < 32; ++k) acc = cadd(acc, cmul(sU[k][i], sU[k][j]));
        gij = acc;
        sG[i][j] = acc;
    }
    __syncthreads();

    // ---- M = (1+lambda)I - G ; Inv = I ----
    sM[i][j] = make_float2(((i == j) ? (1.0f + LAMBDA_P) : 0.0f) - gij.x, -gij.y);
    sI[i][j] = make_float2((i == j) ? 1.0f : 0.0f, 0.0f);
    __syncthreads();

    // ---- Gauss-Jordan inversion with partial pivoting ----
    for (int k = 0; k < 32; ++k) {
        if (tid == 0) {
            int piv = k;
            float best = fabsf(sM[k][k].x) + fabsf(sM[k][k].y);
            for (int r = k + 1; r < 32; ++r) {
                float m = fabsf(sM[r][k].x) + fabsf(sM[r][k].y);
                if (m > best) { best = m; piv = r; }
            }
            sPiv = piv;
        }
        __syncthreads();
        int piv = sPiv;
        if (piv != k && tid < 32) {
            float2 t = sM[k][tid]; sM[k][tid] = sM[piv][tid]; sM[piv][tid] = t;
            t = sI[k][tid]; sI[k][tid] = sI[piv][tid]; sI[piv][tid] = t;
        }
        __syncthreads();
        float2 pv = sM[k][k];
        float den = pv.x * pv.x + pv.y * pv.y;
        float2 pinv = make_float2(pv.x / den, -pv.y / den);
        __syncthreads();
        if (tid < 32) {
            sM[k][tid] = cmul(sM[k][tid], pinv);
            sI[k][tid] = cmul(sI[k][tid], pinv);
        }
        __syncthreads();
        float2 f = sM[i][k];
        float2 mk = sM[k][j];
        float2 ik = sI[k][j];
        __syncthreads();
        if (i != k) {
            sM[i][j] = csub(sM[i][j], cmul(f, mk));
            sI[i][j] = csub(sI[i][j], cmul(f, ik));
        }
        __syncthreads();
    }

    // ---- mix = G @ Inv  -> sM ----
    {
        float2 acc = make_float2(0.0f, 0.0f);
        for (int k = 0; k < 32; ++k) acc = cadd(acc, cmul(sG[i][k], sI[k][j]));
        __syncthreads();
        sM[i][j] = acc;
    }
    __syncthreads();

    // ---- S[i][j] = T2[j][i] = -kappa_j*(0.5*delta_ij + mix[j][i])*kappa_i ----
    {
        float2 mji = sM[j][i];
        float coef = -kappa[j] * kappa[i];
        float sr = coef * (((i == j) ? 0.5f : 0.0f) + mji.x);
        float si = coef * mji.y;
        // W = [[Sr, Si], [-Si, Sr]]  row-major 64x64
        Wg[i * 64 + j] = sr;
        Wg[i * 64 + 32 + j] = si;
        Wg[(32 + i) * 64 + j] = -si;
        Wg[(32 + i) * 64 + 32 + j] = sr;
    }
}

// ===========================================================================
// Kernel 2: RK4 integration. 1 wave (32 threads) per 16 batch rows.
//
// LDS: sX2  : 16x34 v2f tile, row r = [Ar(0..31) | Ai(0..31)], stride 34 v2f
//      sWp  : packed B fragments: sWp[(2k+g)*64 + c] = (W[4k+2g][c], W[4k+2g+1][c])
// A-fragment (16x4, f32): lane<16 -> row=l15, K={4k,4k+1}; lane>=16 -> K={4k+2,4k+3}
//   == single v2f load sX2[l15*34 + 2k + g].
// B-fragment (4x16, f32): vgpr0 rows {4k,4k+2}, vgpr1 rows {4k+1,4k+3}
//   == single v2f load sWp[(2k+g)*64 + 16n + l15].
// D layout ownership: lane holds (rows rbase..rbase+7) x modes {l15, 16+l15},
// re in tiles n=0/1, im in tiles n=2/3 -> whole RK4 stays in registers.
// ===========================================================================
#define XSTR2 34  // v2f stride of one tile row (68 floats, bank-conflict pad)

__device__ __forceinline__ void gemm_tile(const v2f* __restrict__ sX2,
                                          const v2f* __restrict__ sWp,
                                          int l15, int g, v8f acc[4]) {
    __syncthreads();  // stage tile fully written
#pragma unroll
    for (int k = 0; k < 16; ++k) {
        v2f a = sX2[l15 * XSTR2 + 2 * k + g];
#pragma unroll
        for (int n = 0; n < 4; ++n) {
            v2f b = sWp[(2 * k + g) * 64 + 16 * n + l15];
            acc[n] = __builtin_amdgcn_wmma_f32_16x16x4_f32(false, a, false, b,
                                                           (short)0, acc[n], false, false);
        }
    }
    __syncthreads();  // GEMM loads done; tile may be overwritten
}

__global__ __launch_bounds__(32) void rk4_kernel(const float* __restrict__ A0r,
                                                 const float* __restrict__ A0i,
                                                 const float* __restrict__ omega,
                                                 const float* __restrict__ nonlin,
                                                 const float* __restrict__ Wg,
                                                 float2* __restrict__ out2, int B) {
    __shared__ v2f sWp[32 * 64];   // 16 KB packed B fragments
    __shared__ v2f sX2[16 * XSTR2];

    const int lane = threadIdx.x;
    const int l15 = lane & 15;
    const int g = lane >> 4;        // 0: lanes 0-15, 1: lanes 16-31
    const int rbase = g * 8;        // D-fragment row base
    const int m1 = l15;             // owned mode 1
    const int m2 = 16 + l15;        // owned mode 2
    const int gRow0 = blockIdx.x * 16 + rbase;

    float* sXf = (float*)sX2;

    __builtin_prefetch(Wg, 0, 3);
    // pack W into B-fragment order
    for (int idx = lane; idx < 32 * 64; idx += 32) {
        int kg = idx >> 6;   // 2k+g
        int c = idx & 63;
        v2f w;
        w[0] = Wg[(2 * kg) * 64 + c];
        w[1] = Wg[(2 * kg + 1) * 64 + c];
        sWp[kg * 64 + c] = w;
    }

    const float nl2 = nonlin[0] * nonlin[0];
    const float om1 = omega[m1];
    const float om2 = omega[m2];

    // ---- A_init (registers, D-fragment ownership) + t=0 trajectory ----
    float aR1[8], aI1[8], aR2[8], aI2[8];
#pragma unroll
    for (int r = 0; r < 8; ++r) {
        long grow = gRow0 + r;
        aR1[r] = (m1 < IN_MODES) ? A0r[grow * IN_MODES + m1] : 1.0f;
        aI1[r] = (m1 < IN_MODES) ? A0i[grow * IN_MODES + m1] : 0.0f;
        aR2[r] = (m2 < IN_MODES) ? A0r[grow * IN_MODES + m2] : 1.0f;
        aI2[r] = (m2 < IN_MODES) ? A0i[grow * IN_MODES + m2] : 0.0f;
        out2[grow * N_MODES + m1] = make_float2(aR1[r], aI1[r]);
        out2[grow * N_MODES + m2] = make_float2(aR2[r], aI2[r]);
    }

    const float dt = 1.0f / (float)(EVAL_PTS - 1);
    const v8f vz = {0.0f, 0.0f, 0.0f, 0.0f, 0.0f, 0.0f, 0.0f, 0.0f};

    float xR1[8], xI1[8], xR2[8], xI2[8];   // current stage input X
    float cR1[8], cI1[8], cR2[8], cI2[8];   // RK4 accumulators
    float fR1[8], fI1[8], fR2[8], fI2[8];   // stage derivative

    for (int t = 0; t < EVAL_PTS - 1; ++t) {
        // ---------------- stage 1: X = A ----------------
#pragma unroll
        for (int r = 0; r < 8; ++r) {
            xR1[r] = aR1[r]; xI1[r] = aI1[r];
            xR2[r] = aR2[r]; xI2[r] = aI2[r];
            int row = rbase + r;
            sXf[row * 68 + m1] = xR1[r]; sXf[row * 68 + 32 + m1] = xI1[r];
            sXf[row * 68 + m2] = xR2[r]; sXf[row * 68 + 32 + m2] = xI2[r];
        }
#pragma unroll 1
        for (int stage = 0; stage < 4; ++stage) {
            v8f acc[4] = {vz, vz, vz, vz};
            gemm_tile(sX2, sWp, l15, g, acc);
            // f = GEMM + i*(om + nl2*|X|^2).*X  (all registers)
#pragma unroll
            for (int r = 0; r < 8; ++r) {
                float s1 = om1 + nl2 * (xR1[r] * xR1[r] + xI1[r] * xI1[r]);
                float s2 = om2 + nl2 * (xR2[r] * xR2[r] + xI2[r] * xI2[r]);
                fR1[r] = acc[0][r] - s1 * xI1[r];
                fI1[r] = acc[2][r] + s1 * xR1[r];
                fR2[r] = acc[1][r] - s2 * xI2[r];
                fI2[r] = acc[3][r] + s2 * xR2[r];
            }
            if (stage == 0) {
#pragma unroll
                for (int r = 0; r < 8; ++r) { cR1[r] = fR1[r]; cI1[r] = fI1[r]; cR2[r] = fR2[r]; cI2[r] = fI2[r]; }
            } else if (stage == 3) {
#pragma unroll
                for (int r = 0; r < 8; ++r) { cR1[r] += fR1[r]; cI1[r] += fI1[r]; cR2[r] += fR2[r]; cI2[r] += fI2[r]; }
            } else {
#pragma unroll
                for (int r = 0; r < 8; ++r) { cR1[r] += 2.0f * fR1[r]; cI1[r] += 2.0f * fI1[r]; cR2[r] += 2.0f * fR2[r]; cI2[r] += 2.0f * fI2[r]; }
            }
            if (stage < 3) {
                const float h = (stage == 2) ? dt : (0.5f * dt);
#pragma unroll
                for (int r = 0; r < 8; ++r) {
                    xR1[r] = aR1[r] + h * fR1[r]; xI1[r] = aI1[r] + h * fI1[r];
                    xR2[r] = aR2[r] + h * fR2[r]; xI2[r] = aI2[r] + h * fI2[r];
                    int row = rbase + r;
                    sXf[row * 68 + m1] = xR1[r]; sXf[row * 68 + 32 + m1] = xI1[r];
                    sXf[row * 68 + m2] = xR2[r]; sXf[row * 68 + 32 + m2] = xI2[r];
                }
            }
        }
        // ---------------- state update + trajectory store ----------------
        const long base = (long)(t + 1) * B;
#pragma unroll
        for (int r = 0; r < 8; ++r) {
            aR1[r] += (dt / 6.0f) * cR1[r]; aI1[r] += (dt / 6.0f) * cI1[r];
            aR2[r] += (dt / 6.0f) * cR2[r]; aI2[r] += (dt / 6.0f) * cI2[r];
            long grow = base + gRow0 + r;
            out2[grow * N_MODES + m1] = make_float2(aR1[r], aI1[r]);
            out2[grow * N_MODES + m2] = make_float2(aR2[r], aI2[r]);
        }
    }
}

// ===========================================================================
extern "C" void kernel_launch(void* const* d_in, const int* in_sizes, int n_in,
                              void* d_out, int out_size, void* d_ws, size_t ws_size,
                              hipStream_t stream) {
    const float* A0r = (const float*)d_in[0];
    const float* A0i = (const float*)d_in[1];
    const float* omega = (const float*)d_in[2];
    const float* kappa = (const float*)d_in[3];
    const float* nonlin = (const float*)d_in[4];
    const float* params = (const float*)d_in[5];

    float* Wg = (float*)d_ws;  // 64*64 floats = 16 KB
    const int B = in_sizes[0] / IN_MODES;

    setup_kernel<<<dim3(1), dim3(1024), 0, stream>>>(params, kappa, Wg);
    rk4_kernel<<<dim3(B / 16), dim3(32), 0, stream>>>(A0r, A0i, omega, nonlin, Wg,
                                                      (float2*)d_out, B);
}